// Graph_Processer_Belief_Regulator_43860206027000
// MI455X (gfx1250) — compile-verified
//
#include <hip/hip_runtime.h>
#include <hip/hip_bf16.h>
#include <math.h>

// MI455X (gfx1250) implementation.
// Bandwidth-bound: ~587 MB stored per call -> ~27us floor @ 23.3 TB/s.
// Strategy: 1 block per (b,t) slice, LDS staging, WMMA f32 16x16x4 for the
// two one-hot einsums, coalesced 128-bit non-temporal streaming stores.

typedef __attribute__((ext_vector_type(2))) float v2f;
typedef __attribute__((ext_vector_type(4))) float f4;
typedef __attribute__((ext_vector_type(8))) float v8f;

#define N_OBJ  20
#define NREL   380   // N*(N-1)
#define NTHR   384   // 12 wave32s

__global__ __launch_bounds__(NTHR)
void gp_belief_kernel(const float* __restrict__ state,   // (BT, 20, 8)
                      const float* __restrict__ shape,   // (BT, 20, 4)
                      const float* __restrict__ reli,    // (BT, 380, 3)
                      float* __restrict__ outRel,        // (BT, 380, 20)
                      float* __restrict__ outObj,        // (BT, 20, 7)
                      float* __restrict__ outRecv,       // (BT, 380, 20)
                      float* __restrict__ outSend)       // (BT, 380, 20)
{
    // LDS: 1280 + 1536 + 36864 = ~39.7 KB (well under 320KB/WGP)
    __shared__ float Dmat[N_OBJ][16];               // [state(8) | shape(4) | 0(4)]
    __shared__ float maskL[NTHR];
    __shared__ __align__(16) float scratch[NTHR * 24]; // recv[384*12] | send[384*12]; reused as relrow[380*20]

    const int    tid = threadIdx.x;
    const int    bt  = blockIdx.x;
    const size_t sbt = (size_t)bt;

    // Early prefetch of this slice's relation_info (global_prefetch_b8)
    if (tid < NREL) __builtin_prefetch(reli + (sbt * NREL + tid) * 3, 0, 3);

    // ---- Phase 0: stage D = [state | shape | 0] (20 x 16) in LDS ----
    if (tid < N_OBJ * 16) {
        int n = tid >> 4, c = tid & 15;
        float v = 0.f;
        if (c < 8)       v = state[(sbt * N_OBJ + n) * 8 + c];
        else if (c < 12) v = shape[(sbt * N_OBJ + n) * 4 + (c - 8)];
        Dmat[n][c] = v;
    }
    __syncthreads();

    // ---- Phase 1: adjacency mask per relation ----
    if (tid < NREL) {
        int i = tid / 19, t = tid - i * 19;
        int j = (t < i) ? t : t + 1;
        float dx = Dmat[i][0] - Dmat[j][0];
        float dy = Dmat[i][1] - Dmat[j][1];
        float d2 = dx * dx + dy * dy;
        float d  = sqrtf(d2);
        maskL[tid] = (d < 0.35f && d2 > 0.f) ? 1.f : 0.f;
    }
    __syncthreads();

    // ---- Phase 2: einsum (380x20)x(20x12) via V_WMMA_F32_16X16X4_F32 ----
    // A (16x4 f32): lanes 0-15 hold K=k0,k0+1 ; lanes 16-31 hold K=k0+2,k0+3.
    // B (4x16 f32): VGPR0 = rows k0 / k0+2 ; VGPR1 = rows k0+1 / k0+3.
    // C/D (16x16 f32, 8 VGPRs): lane<16 -> M=v ; lane>=16 -> M=v+8 ; N=lane&15.
    {
        const int lane = tid & 31;
        const int wave = tid >> 5;       // 0..11
        const int m16  = lane & 15;
        const int kh   = lane >> 4;      // 0 or 1

        for (int tile = wave; tile < 24; tile += 12) {   // uniform per wave: 2 iters
            const int rbase = tile * 16;
            const int r     = rbase + m16;
            const int rc    = (r < NREL) ? r : 0;
            const float mk  = (r < NREL) ? maskL[r] : 0.f;
            const int i     = rc / 19;
            const int t     = rc - i * 19;
            const int j     = (t < i) ? t : t + 1;

            v8f cR = {};   // receivers accumulator
            v8f cS = {};   // senders accumulator
#pragma unroll
            for (int kc = 0; kc < 5; ++kc) {             // K = 20 = 5 * 4
                const int k0 = kc * 4 + kh * 2;
                v2f aR, aS, bb;
                aR.x = (i == k0)     ? mk : 0.f;
                aR.y = (i == k0 + 1) ? mk : 0.f;
                aS.x = (j == k0)     ? mk : 0.f;
                aS.y = (j == k0 + 1) ? mk : 0.f;
                bb.x = Dmat[k0][m16];
                bb.y = Dmat[k0 + 1][m16];
                cR = __builtin_amdgcn_wmma_f32_16x16x4_f32(false, aR, false, bb,
                                                           (short)0, cR, false, false);
                cS = __builtin_amdgcn_wmma_f32_16x16x4_f32(false, aS, false, bb,
                                                           (short)0, cS, false, false);
            }
#pragma unroll
            for (int v = 0; v < 8; ++v) {
                int row = rbase + kh * 8 + v;
                if (row < NREL && m16 < 12) {
                    scratch[row * 12 + m16]             = cR[v];   // receivers
                    scratch[NTHR * 12 + row * 12 + m16] = cS[v];   // senders
                }
            }
        }
    }
    __syncthreads();

    // ---- Phase 3a: pull my relation's rows into registers ----
    float rv[12], sd[12], rinfo[3];
    if (tid < NREL) {
#pragma unroll
        for (int k = 0; k < 12; ++k) {
            rv[k] = scratch[tid * 12 + k];
            sd[k] = scratch[NTHR * 12 + tid * 12 + k];
        }
        const float* rp = reli + (sbt * NREL + tid) * 3;
        rinfo[0] = rp[0]; rinfo[1] = rp[1]; rinfo[2] = rp[2];
    }
    __syncthreads();   // scratch now reusable as relrow

    // ---- Phase 3b: per-relation feature row (20 floats) into LDS ----
    if (tid < NREL) {
        float ra = rv[2];
        float ca = cosf(ra), sa = sinf(ra);      // basis rows: [ca,sa], [-sa,ca]
        float dx  = sd[0] - rv[0], dy  = sd[1] - rv[1];
        float dvx = sd[3] - rv[3], dvy = sd[4] - rv[4];
        float cx  = sd[6],          cy = sd[7];
        float adiff = sd[2] - rv[2];
        float adv   = sd[5] - rv[5];
        float* row = &scratch[tid * 20];
        row[0]  = rinfo[0]; row[1] = rinfo[1]; row[2] = rinfo[2];
        row[3]  =  ca * dx  + sa * dy;
        row[4]  = -sa * dx  + ca * dy;
        row[5]  =  ca * dvx + sa * dvy;
        row[6]  = -sa * dvx + ca * dvy;
        row[7]  =  ca * cx  + sa * cy;
        row[8]  = -sa * cx  + ca * cy;
        row[9]  = sinf(2.f * adiff);
        row[10] = cosf(2.f * adiff);
        row[11] = adv;
        row[12] = rv[8];  row[13] = rv[9];  row[14] = rv[10]; row[15] = rv[11];
        row[16] = sd[8];  row[17] = sd[9];  row[18] = sd[10]; row[19] = sd[11];
    }

    // ---- obj_data (20 x 7) ----
    if (tid < N_OBJ) {
        float ang = Dmat[tid][2], vx = Dmat[tid][3], vy = Dmat[tid][4];
        float ca = cosf(ang), sa = sinf(ang);
        float* o = outObj + sbt * (N_OBJ * 7) + tid * 7;
        o[0] =  ca * vx + sa * vy;
        o[1] = -sa * vx + ca * vy;
        o[2] = Dmat[tid][5];
        o[3] = Dmat[tid][8]; o[4] = Dmat[tid][9];
        o[5] = Dmat[tid][10]; o[6] = Dmat[tid][11];
    }
    __syncthreads();

    // ---- Phase 4: coalesced 128-bit non-temporal streaming stores ----
    // rel_data: 380*20 = 7600 floats = 1900 f4
    {
        const f4* src = reinterpret_cast<const f4*>(scratch);
        f4* dst = reinterpret_cast<f4*>(outRel + sbt * (NREL * 20));
        for (int g = tid; g < 1900; g += NTHR)
            __builtin_nontemporal_store(src[g], &dst[g]);
    }
    // receiver/sender relation matrices: mask * one-hot, generated in registers
    {
        f4* dR = reinterpret_cast<f4*>(outRecv + sbt * (NREL * 20));
        f4* dS = reinterpret_cast<f4*>(outSend + sbt * (NREL * 20));
        for (int g = tid; g < 1900; g += NTHR) {
            int e0 = g * 4;
            int r  = e0 / 20;
            int n0 = e0 - r * 20;
            float mk = maskL[r];
            int i = r / 19, t = r - i * 19;
            int j = (t < i) ? t : t + 1;
            f4 vr, vs;
            vr.x = (n0     == i) ? mk : 0.f;  vs.x = (n0     == j) ? mk : 0.f;
            vr.y = (n0 + 1 == i) ? mk : 0.f;  vs.y = (n0 + 1 == j) ? mk : 0.f;
            vr.z = (n0 + 2 == i) ? mk : 0.f;  vs.z = (n0 + 2 == j) ? mk : 0.f;
            vr.w = (n0 + 3 == i) ? mk : 0.f;  vs.w = (n0 + 3 == j) ? mk : 0.f;
            __builtin_nontemporal_store(vr, &dR[g]);
            __builtin_nontemporal_store(vs, &dS[g]);
        }
    }
}

extern "C" void kernel_launch(void* const* d_in, const int* in_sizes, int n_in,
                              void* d_out, int out_size, void* d_ws, size_t ws_size,
                              hipStream_t stream) {
    const float* st = (const float*)d_in[0];   // objects_state (B,T,20,8) f32
    const float* sh = (const float*)d_in[1];   // objects_shape (B,T,20,4) f32
    const float* ri = (const float*)d_in[2];   // relation_info (B,T,380,3) f32
    float* out = (float*)d_out;

    const int BT = in_sizes[0] / (N_OBJ * 8);  // 6400
    const size_t relSz = (size_t)BT * NREL * 20;
    const size_t objSz = (size_t)BT * N_OBJ * 7;

    float* oRel  = out;                 // rel_data
    float* oObj  = out + relSz;         // obj_data
    float* oRecv = oObj + objSz;        // receiver_relations
    float* oSend = oRecv + relSz;       // sender_relations

    gp_belief_kernel<<<BT, NTHR, 0, stream>>>(st, sh, ri, oRel, oObj, oRecv, oSend);
}